// ChamferLossP_33646773796927
// MI455X (gfx1250) — compile-verified
//
#include <hip/hip_runtime.h>
#include <math.h>

typedef __attribute__((ext_vector_type(2))) float v2f;
typedef __attribute__((ext_vector_type(8))) float v8f;

#define BATCH 8
#define NPTS 4096
#define WAVES_PER_BLOCK 8
#define ROWS_PER_WAVE 16
#define ROWS_PER_BLOCK (WAVES_PER_BLOCK * ROWS_PER_WAVE) /* 128 */
#define BLOCKS_PER_SET (NPTS / ROWS_PER_BLOCK)           /* 32  */

__global__ void chamfer_zero_acc(float* acc) {
    if (threadIdx.x < 2 * BATCH) acc[threadIdx.x] = 0.0f;
}

__global__ __launch_bounds__(256)
void chamfer_nn_kernel(const float* __restrict__ x,
                       const float* __restrict__ y,
                       float* __restrict__ acc) {
    // Search set staged as float4 per point {bx, by, bz, ||b||^2}: exactly 64KB.
    __shared__ float sY[NPTS * 4];

    const int b   = blockIdx.y;
    const int dir = blockIdx.z;
    const float* __restrict__ A  = (dir == 0) ? (x + (size_t)b * NPTS * 3)
                                              : (y + (size_t)b * NPTS * 3);
    const float* __restrict__ Bp = (dir == 0) ? (y + (size_t)b * NPTS * 3)
                                              : (x + (size_t)b * NPTS * 3);

    // Cooperative fill: point t -> {bx, by, bz, yy}. The yy term rides in the
    // K=3 slot of the WMMA B operand so the matrix unit computes the full
    // comparison key dot' = x.y - 0.5*yy  (argmin d == argmax dot').
    for (int t = threadIdx.x; t < NPTS; t += 256) {
        float4 v;
        v.x = Bp[t * 3 + 0];
        v.y = Bp[t * 3 + 1];
        v.z = Bp[t * 3 + 2];
        v.w = v.x * v.x + v.y * v.y + v.z * v.z;
        *(float4*)&sY[t * 4] = v;
    }
    __syncthreads();

    const int lane = threadIdx.x & 31;
    const int wave = threadIdx.x >> 5;
    const int h    = lane >> 4;   // K-half: lanes 0-15 -> K{0,1}, 16-31 -> K{2,3}
    const int lrow = lane & 15;
    const int i0   = blockIdx.x * ROWS_PER_BLOCK + wave * ROWS_PER_WAVE;

    // A operand (16x4 f32): K = {x0,x1,x2,-0.5}. Loop invariant.
    const int r = i0 + lrow;
    v2f a;
    if (h == 0) { a.x = A[r * 3 + 0]; a.y = A[r * 3 + 1]; }
    else        { a.x = A[r * 3 + 2]; a.y = -0.5f; }

    float runMax[8];
    int   runIdx[8];
#pragma unroll
    for (int v = 0; v < 8; ++v) { runMax[v] = -3.4e38f; runIdx[v] = 0; }

    for (int j0 = 0; j0 < NPTS; j0 += 16) {
        const int c = j0 + lrow;                 // this lane's column (all 8 slots)
        // B operand (4x16 f32): {Y[c][2h], Y[c][2h+1]}; half1 = {z, yy}.
        v2f bv = *(const v2f*)&sY[c * 4 + h * 2];

        v8f cz = {0.f, 0.f, 0.f, 0.f, 0.f, 0.f, 0.f, 0.f};
        v8f dot = __builtin_amdgcn_wmma_f32_16x16x4_f32(
            false, a, false, bv, (short)0, cz, false, false);

#pragma unroll
        for (int v = 0; v < 8; ++v) {
            bool gt = dot[v] > runMax[v];        // strict: first occurrence wins
            runMax[v] = gt ? dot[v] : runMax[v];
            runIdx[v] = gt ? c      : runIdx[v];
        }
    }

    // Row-wise argmax: butterfly inside each 16-lane half, index tie-break
    // (smaller index wins -> argmin first-occurrence semantics).
#pragma unroll
    for (int s = 1; s <= 8; s <<= 1) {
#pragma unroll
        for (int v = 0; v < 8; ++v) {
            float om = __shfl_xor(runMax[v], s, 32);
            int   oi = __shfl_xor(runIdx[v], s, 32);
            bool take = (om > runMax[v]) || (om == runMax[v] && oi < runIdx[v]);
            runMax[v] = take ? om : runMax[v];
            runIdx[v] = take ? oi : runIdx[v];
        }
    }

    // Lane 0 holds results for rows i0+0..7, lane 16 for rows i0+8..15.
    float s5 = 0.0f;
    if (lrow == 0) {
#pragma unroll
        for (int v = 0; v < 8; ++v) {
            int row = i0 + h * 8 + v;
            int nn  = runIdx[v];
            float dx = fabsf(A[row * 3 + 0] - sY[nn * 4 + 0]);
            float dy = fabsf(A[row * 3 + 1] - sY[nn * 4 + 1]);
            float dz = fabsf(A[row * 3 + 2] - sY[nn * 4 + 2]);
            float dx2 = dx * dx, dy2 = dy * dy, dz2 = dz * dz;
            s5 += dx2 * dx2 * dx + dy2 * dy2 * dy + dz2 * dz2 * dz; // |d|^5 sum
        }
    }
#pragma unroll
    for (int s = 1; s <= 16; s <<= 1) s5 += __shfl_xor(s5, s, 32);
    if (lane == 0) atomicAdd(&acc[dir * BATCH + b], s5);
}

__global__ void chamfer_finalize(const float* __restrict__ acc,
                                 float* __restrict__ out) {
    if (threadIdx.x == 0) {
        float m = 0.0f;
        for (int b = 0; b < BATCH; ++b)
            m += powf(acc[b], 0.2f) + powf(acc[BATCH + b], 0.2f);
        out[0] = m * (1.0f / BATCH);
    }
}

extern "C" void kernel_launch(void* const* d_in, const int* in_sizes, int n_in,
                              void* d_out, int out_size, void* d_ws, size_t ws_size,
                              hipStream_t stream) {
    (void)in_sizes; (void)n_in; (void)out_size; (void)ws_size;
    const float* x = (const float*)d_in[0];   // [8, 4096, 3] f32
    const float* y = (const float*)d_in[1];   // [8, 4096, 3] f32
    float* acc = (float*)d_ws;                // 16 floats: [dir][batch] sum |diff|^5
    float* out = (float*)d_out;               // 1 float

    chamfer_zero_acc<<<1, 32, 0, stream>>>(acc);

    dim3 grid(BLOCKS_PER_SET, BATCH, 2);
    dim3 block(256);
    chamfer_nn_kernel<<<grid, block, 0, stream>>>(x, y, acc);

    chamfer_finalize<<<1, 32, 0, stream>>>(acc, out);
}